// SparsePropMaxPool_33638183862771
// MI455X (gfx1250) — compile-verified
//
#include <hip/hip_runtime.h>
#include <stdint.h>

typedef float v4f __attribute__((ext_vector_type(4)));

#define ROWS_PER_BLOCK 8
#define LEVELS 7   // sparse-table levels k=0..6 (windows 1..64)

// Active-cell predicate derived from the reference's scatter pattern:
//   d = e-s:  d in [0,15]                      -> active for any s
//             d in {17,19,...,31} and s even   -> active (scale 1)
//             d in {35,39,...,63} and s%4==0   -> active (scale 2)
__device__ __forceinline__ bool is_active(int s, int e) {
    int d = e - s;
    if (d < 0) return false;
    if (d <= 15) return true;
    if (d <= 31) return (d >= 17) && (d & 1) && ((s & 1) == 0);
    return (d >= 35) && (((d - 35) & 3) == 0) && ((s & 3) == 0);
}

__global__ __launch_bounds__(256) void sppool_kernel(const float* __restrict__ x,
                                                     float* __restrict__ out,
                                                     int nRows, int H) {
    __shared__ float tbl[ROWS_PER_BLOCK][LEVELS][64];

    const int lane = threadIdx.x & 31;
    const int w    = threadIdx.x >> 5;
    const int row  = blockIdx.x * ROWS_PER_BLOCK + w;

    float* const mapOut  = out;
    float* const maskOut = out + (size_t)nRows * 4096;

    // ---- CDNA5 async global -> LDS staging of this wave's x row ----
    // Each lane copies 8 bytes, x[row*64 + 2*lane .. +1] in global memory, to tbl[w][0][2*lane] in LDS
    {
        const float* g = x + (size_t)row * 64 + lane * 2;
        float* l = &tbl[w][0][lane * 2];
        unsigned lds_off = (unsigned)(uintptr_t)l;          // low 32 bits = LDS byte offset
        unsigned long long ga = (unsigned long long)(uintptr_t)g;
        asm volatile("global_load_async_to_lds_b64 %0, %1, off"
                     :: "v"(lds_off), "v"(ga) : "memory");
        asm volatile("s_wait_asynccnt 0" ::: "memory");     // ASYNCcnt: data landed in LDS
    }
    __syncthreads();

    // ---- build doubling max table: tbl[k][s] = max(x[s .. min(s+2^k-1, 63)]) ----
    for (int k = 1; k < LEVELS; ++k) {
        const int off = 1 << (k - 1);
        #pragma unroll
        for (int j = 0; j < 2; ++j) {
            int s  = lane + 32 * j;
            int s2 = s + off; if (s2 > 63) s2 = 63;
            tbl[w][k][s] = fmaxf(tbl[w][k - 1][s], tbl[w][k - 1][s2]);
        }
        __syncthreads();
    }

    // ---- stream the 64x64 tile: 32 iterations of coalesced float4 NT stores ----
    float* base = mapOut + (size_t)row * 4096;
    #pragma unroll 8
    for (int t = 0; t < 32; ++t) {
        const int idx4 = t * 32 + lane;      // float4 index within tile
        const int s    = idx4 >> 4;          // tile row (16 float4 per 64-col row)
        const int c0   = (idx4 & 15) * 4;    // first column of this float4
        v4f v;
        #pragma unroll
        for (int j = 0; j < 4; ++j) {
            const int e = c0 + j;
            float val = 0.0f;
            if (is_active(s, e)) {
                const int wl = e - s + 1;                    // window length 1..64
                const int k  = 31 - __builtin_clz((unsigned)wl);
                val = fmaxf(tbl[w][k][s], tbl[w][k][e - (1 << k) + 1]);
            }
            v[j] = val;
        }
        __builtin_nontemporal_store(v, (v4f*)(base + idx4 * 4)); // b128 NT store
    }

    // ---- one wave per batch writes the (batch-invariant) mask tile ----
    if ((row % H) == 0) {
        float* mbase = maskOut + (size_t)(row / H) * 4096;
        #pragma unroll 8
        for (int t = 0; t < 32; ++t) {
            const int idx4 = t * 32 + lane;
            const int s    = idx4 >> 4;
            const int c0   = (idx4 & 15) * 4;
            v4f v;
            #pragma unroll
            for (int j = 0; j < 4; ++j)
                v[j] = is_active(s, c0 + j) ? 1.0f : 0.0f;
            __builtin_nontemporal_store(v, (v4f*)(mbase + idx4 * 4));
        }
    }
}

extern "C" void kernel_launch(void* const* d_in, const int* in_sizes, int n_in,
                              void* d_out, int out_size, void* d_ws, size_t ws_size,
                              hipStream_t stream) {
    (void)n_in; (void)out_size; (void)d_ws; (void)ws_size;
    const float* x = (const float*)d_in[0];
    float* out = (float*)d_out;

    const int N = 64;                       // clip dimension (reference: x is (32,512,64))
    const int H = 512;
    const int nRows = in_sizes[0] / N;      // 32*512 = 16384 rows, divisible by 8

    dim3 block(256);
    dim3 grid(nRows / ROWS_PER_BLOCK);      // 2048 blocks, one wave32 per row
    hipLaunchKernelGGL(sppool_kernel, grid, block, 0, stream, x, out, nRows, H);
}